// MinkConvBNRelu_51144470561084
// MI455X (gfx1250) — compile-verified
//
#include <hip/hip_runtime.h>
#include <hip/hip_bf16.h>

// ---------------------------------------------------------------------------
// MinkowskiConv(3x3x3, C32->C32) + BatchNorm + ReLU on MI455X (gfx1250).
// Per-offset tiled GEMM on V_WMMA_F32_16X16X4_F32 (fp32 matrix core) with the
// 32x32 weight tile staged in LDS (pair-swizzled for ds_load_b64 B fragments),
// scatter-add via global_atomic_add_f32, two-pass BN + ReLU on top.
// ---------------------------------------------------------------------------

#define NVOX 120000
#define CIN  32
#define COUT 32
#define KVOL 27
#define TILES ((NVOX + 15) / 16)        // 7500, exact
#define WAVES_PER_BLOCK 8
#define BN_EPS 1e-5f

typedef __attribute__((ext_vector_type(2))) float v2f;
typedef __attribute__((ext_vector_type(8))) float v8f;

// ---------------------------------------------------------------------------
// Zero the accumulator + stats region of the workspace (graph-replay safe).
// ---------------------------------------------------------------------------
__global__ void __launch_bounds__(256) mink_zero_kernel(float* __restrict__ acc,
                                                        float* __restrict__ stats) {
    long i = (long)blockIdx.x * blockDim.x + threadIdx.x;
    if (i < (long)NVOX * COUT) acc[i] = 0.0f;
    if (i < 64) stats[i] = 0.0f;
}

// ---------------------------------------------------------------------------
// Sparse conv: blockIdx.y = kernel offset k, one wave = one 16-point tile.
// Each wave computes gathered[16x32] * W[k][32x32] with 16
// v_wmma_f32_16x16x4_f32 (8 K-steps x 2 N-tiles), B fragments from LDS.
//
// WMMA f32 lane layout (ISA 7.12.2):
//   A 16x4:  lane l<16 -> {A[l][4s+0], A[l][4s+1]}, lane l+16 -> K=+2,+3
//   B 4x16:  lane l<16 -> {B[4s+0][l], B[4s+1][l]}, lane l+16 -> K=+2,+3
//   C/D:     VGPR v, lane l<16 -> D[v][l]; lane l+16 -> D[v+8][l-16]
//
// LDS layout: bsh[2*(s*64 + h*32 + n) + {0,1}] = { W[4s+2h][n], W[4s+2h+1][n] }
// so each lane's B fragment is one aligned 8-byte DS load.
// ---------------------------------------------------------------------------
__global__ void __launch_bounds__(256) mink_conv_wmma_kernel(
    const float* __restrict__ feats,          // [NVOX, CIN]
    const float* __restrict__ W,              // [KVOL, CIN, COUT]
    const int*   __restrict__ in_idx,         // [KVOL, NVOX]
    const int*   __restrict__ out_idx,        // [KVOL, NVOX]
    const unsigned char* __restrict__ mask,   // [KVOL, NVOX] (1-byte bool)
    float* __restrict__ acc)                  // [NVOX, COUT] accumulator
{
    __shared__ float bsh[2 * CIN * COUT / 2]; // 1024 floats = 4 KB

    const int k = blockIdx.y;
    const float* wk = W + (long)k * (CIN * COUT);

    // ---- Stage W[k] into LDS, pair-swizzled (all 256 threads, coalesced).
    for (int t = threadIdx.x; t < 512; t += 256) {
        const int s = t >> 6;                 // K-step 0..7
        const int h = (t >> 5) & 1;           // half
        const int n = t & 31;                 // output channel
        const int kr = 4 * s + 2 * h;
        bsh[2 * t + 0] = wk[kr * COUT + n];
        bsh[2 * t + 1] = wk[(kr + 1) * COUT + n];
    }
    __syncthreads();

    const int lane = threadIdx.x & 31;
    const int wave = threadIdx.x >> 5;
    const int tile = blockIdx.x * WAVES_PER_BLOCK + wave;
    if (tile >= TILES) return;                // wave-uniform exit after barrier

    const int p0   = tile * 16;
    const int half = lane >> 4;               // 0: K pair {0,1}; 1: K pair {2,3}
    const int lm   = lane & 15;

    // ---- A operand: this lane's gathered feature row.
    const long kbase  = (long)k * NVOX;
    const long mrow   = kbase + (p0 + lm);
    const bool avalid = mask[mrow] != 0;
    const int  ain    = avalid ? in_idx[mrow] : 0;
    const float* afp  = feats + (long)ain * CIN + 2 * half;

    v8f c0 = {};   // N-tile 0: channels [0,16)
    v8f c1 = {};   // N-tile 1: channels [16,32)

    #pragma unroll
    for (int s = 0; s < 8; ++s) {             // K = 32 as 8 steps of 4
        v2f a;
        if (avalid) {
            a.x = afp[4 * s + 0];
            a.y = afp[4 * s + 1];
        } else {
            a.x = 0.0f; a.y = 0.0f;
        }
        const int bbase = 2 * ((s * 2 + half) * 32 + lm);
        const v2f b0 = *(const v2f*)&bsh[bbase];        // channels [0,16)
        const v2f b1 = *(const v2f*)&bsh[bbase + 32];   // channels [16,32)

        // 8 args: (neg_a, A, neg_b, B, c_mod, C, reuse_a, reuse_b)
        c0 = __builtin_amdgcn_wmma_f32_16x16x4_f32(false, a, false, b0,
                                                   (short)0, c0, false, false);
        c1 = __builtin_amdgcn_wmma_f32_16x16x4_f32(false, a, false, b1,
                                                   (short)0, c1, false, false);
    }

    // ---- Scatter-add D. This lane owns rows m = 8*half + 0..7, which are
    // contiguous: one u64 mask load + two b128 index loads cover all 8 rows.
    const long rbase = kbase + p0 + 8 * half;
    const unsigned long long mb =
        *(const unsigned long long*)(mask + rbase);      // 8 mask bytes
    const int4 oi0 = *(const int4*)(out_idx + rbase);
    const int4 oi1 = *(const int4*)(out_idx + rbase + 4);
    const int orow[8] = {oi0.x, oi0.y, oi0.z, oi0.w,
                         oi1.x, oi1.y, oi1.z, oi1.w};

    #pragma unroll
    for (int v = 0; v < 8; ++v) {
        if ((mb >> (8 * v)) & 0xffULL) {
            const long o = (long)orow[v] * COUT;
            atomicAdd(&acc[o + lm],      c0[v]);
            atomicAdd(&acc[o + 16 + lm], c1[v]);
        }
    }
}

// ---------------------------------------------------------------------------
// Per-channel sum / sum-of-squares: block-local LDS reduction, then one
// atomicAdd per channel per block (64 atomics/block total).
// ---------------------------------------------------------------------------
__global__ void __launch_bounds__(256) mink_stats_kernel(
    const float* __restrict__ acc, float* __restrict__ stats)
{
    __shared__ float ssum[256];
    __shared__ float ssq[256];
    const int c  = threadIdx.x & 31;          // channel
    const int r0 = threadIdx.x >> 5;          // row slot 0..7
    float s = 0.0f, q = 0.0f;
    for (long row = (long)blockIdx.x * 8 + r0; row < NVOX;
         row += (long)gridDim.x * 8) {
        const float x = acc[row * COUT + c];
        s += x;
        q += x * x;
    }
    ssum[threadIdx.x] = s;
    ssq[threadIdx.x]  = q;
    __syncthreads();
    if (threadIdx.x < 32) {
        float ts = 0.0f, tq = 0.0f;
        #pragma unroll
        for (int w = 0; w < 8; ++w) {
            ts += ssum[w * 32 + c];
            tq += ssq[w * 32 + c];
        }
        atomicAdd(&stats[c],      ts);
        atomicAdd(&stats[32 + c], tq);
    }
}

// ---------------------------------------------------------------------------
// y = relu((x - mean) * rsqrt(var + eps) * gamma + beta)
// ---------------------------------------------------------------------------
__global__ void __launch_bounds__(256) mink_bn_relu_kernel(
    const float* __restrict__ acc, const float* __restrict__ stats,
    const float* __restrict__ gamma, const float* __restrict__ beta,
    float* __restrict__ out)
{
    const long i = (long)blockIdx.x * blockDim.x + threadIdx.x;
    if (i >= (long)NVOX * COUT) return;
    const int c = (int)(i & 31);
    const float invN = 1.0f / (float)NVOX;
    const float mean = stats[c] * invN;
    const float var  = stats[32 + c] * invN - mean * mean;
    const float inv  = rsqrtf(var + BN_EPS);
    const float y    = (acc[i] - mean) * inv * gamma[c] + beta[c];
    out[i] = y > 0.0f ? y : 0.0f;
}

// ---------------------------------------------------------------------------
extern "C" void kernel_launch(void* const* d_in, const int* in_sizes, int n_in,
                              void* d_out, int out_size, void* d_ws, size_t ws_size,
                              hipStream_t stream) {
    (void)in_sizes; (void)n_in; (void)out_size; (void)ws_size;

    const float* feats   = (const float*)d_in[0];            // [NVOX, CIN]
    const float* W       = (const float*)d_in[1];            // [KVOL, CIN, COUT]
    const float* gamma   = (const float*)d_in[2];            // [COUT]
    const float* beta    = (const float*)d_in[3];            // [COUT]
    const int*   in_idx  = (const int*)d_in[4];              // [KVOL, NVOX]
    const int*   out_idx = (const int*)d_in[5];              // [KVOL, NVOX]
    const unsigned char* mask = (const unsigned char*)d_in[6]; // [KVOL, NVOX]

    float* acc   = (float*)d_ws;                             // [NVOX, COUT]
    float* stats = acc + (long)NVOX * COUT;                  // [64]
    float* out   = (float*)d_out;

    const int total_elems = NVOX * COUT;                     // 3,840,000
    const int zero_blocks = (total_elems + 255) / 256;
    mink_zero_kernel<<<zero_blocks, 256, 0, stream>>>(acc, stats);

    // k on grid.y so every block is k-uniform and shares one LDS weight tile.
    const int tile_blocks = (TILES + WAVES_PER_BLOCK - 1) / WAVES_PER_BLOCK; // 938
    dim3 conv_grid(tile_blocks, KVOL, 1);
    mink_conv_wmma_kernel<<<conv_grid, 256, 0, stream>>>(
        feats, W, in_idx, out_idx, mask, acc);

    mink_stats_kernel<<<1024, 256, 0, stream>>>(acc, stats);

    mink_bn_relu_kernel<<<zero_blocks, 256, 0, stream>>>(acc, stats, gamma, beta, out);
}